// AutoregressiveConvLSTM_54528904790253
// MI455X (gfx1250) — compile-verified
//
#include <hip/hip_runtime.h>
#include <math.h>

typedef float v8f __attribute__((ext_vector_type(8)));
typedef float v2f __attribute__((ext_vector_type(2)));

#define HDIM 128
#define WDIM 128
#define HW   (HDIM*WDIM)      // 16384 pixels
#define PW   130              // padded row stride (1-pixel zero halo)
#define HPFLOATS (PW*PW*2)    // 33800 floats per padded h buffer
#define NTHREADS 512          // 16 waves (wave32)
#define NWAVES 16
#define PXW 1024              // pixels per wave
#define NIT 32                // strips of 32 pixels per wave
#define TSTEPS 15

// ---- LDS layout (floats) ----
#define OFF_H0  0                       // 33800 : padded h buffer A
#define OFF_H1  HPFLOATS                // 33800 : padded h buffer B
#define OFF_GS  (2*HPFLOATS)            // 4608  : per-wave gate scratch (16 waves * 32px * stride 9)
#define OFF_WT  (OFF_GS + NWAVES*288)   // 160   : staged weights (+ 8B zero slot at +144)
#define OFF_RED (OFF_WT + 160)          // 512   : reduction
#define LDS_FLOATS (OFF_RED + NTHREADS)
#define LDS_BYTES  (LDS_FLOATS*4)       // 291,520 B <= 320KB per workgroup

// staged-weight sub-offsets inside WT
//  [0..8] Win, [9] b_in, [16..87] Wih (tap*8+g), [88..95] b_ih, [96..103] b_hh,
//  [104..139] Wout_y (tap*4+ch*2+o), [140..141] b_out, [144..145] zero slot (8B aligned)

__device__ __forceinline__ float sigf(float x) { return 1.0f / (1.0f + __expf(-x)); }

__global__ void __launch_bounds__(NTHREADS, 1)
clstm_fused_kernel(const float* __restrict__ X,
                   const float* __restrict__ Win,  const float* __restrict__ b_in,
                   const float* __restrict__ Wih,  const float* __restrict__ b_ih,
                   const float* __restrict__ Whh,  const float* __restrict__ b_hh,
                   const float* __restrict__ Wout, const float* __restrict__ b_out,
                   float* __restrict__ out)
{
  extern __shared__ float lds[];
  const int tid  = threadIdx.x;
  const int w    = tid >> 5;      // wave id 0..15
  const int l    = tid & 31;      // lane id
  const int half = l >> 4;        // selects K-pair within a WMMA chunk
  const int n    = l & 15;        // B/D column (gate index when < 8)
  const int b    = blockIdx.x;    // batch image

  float* WT = lds + OFF_WT;

  // ---- stage small weights into LDS ----
  if (tid < 9)  WT[tid]      = Win[tid];
  if (tid == 9) WT[9]        = b_in[0];
  if (tid < 72) WT[16 + tid] = Wih[tid];
  if (tid < 8)  WT[88 + tid] = b_ih[tid];
  if (tid < 8)  WT[96 + tid] = b_hh[tid];
  if (tid < 36) {                                   // Wout_y = Wout[:,:,:2,:]
    int tap = tid >> 2, rem = tid & 3, ch = rem >> 1, o = rem & 1;
    WT[104 + tid] = Wout[(tap*4 + ch)*2 + o];
  }
  if (tid < 2)  WT[140 + tid] = b_out[tid];
  if (tid < 2)  WT[144 + tid] = 0.0f;               // zero slot for padded-K A fragment

  // ---- zero BOTH padded h buffers (halo stays zero forever; writes are interior-only) ----
  for (int i = tid; i < 2*HPFLOATS; i += NTHREADS) lds[OFF_H0 + i] = 0.0f;

  // ---- preload Whh B-matrix fragments (5 chunks of K=4; K = 2*tap + ch, K>=18 -> 0) ----
  v2f barr[5];
#pragma unroll
  for (int j = 0; j < 5; ++j) {
    int k0 = 4*j + 2*half;
    float bx = (k0     < 18 && n < 8) ? Whh[k0*8 + n]       : 0.0f;
    float by = (k0 + 1 < 18 && n < 8) ? Whh[(k0 + 1)*8 + n] : 0.0f;
    barr[j].x = bx; barr[j].y = by;
  }
  // per-lane A-gather relative offsets (floats) in the padded buffer
  int rel[5];
  const bool ok4 = (half == 0);       // j==4,half==1 -> K=18/19 zero padding
#pragma unroll
  for (int j = 0; j < 5; ++j) {
    int tap = 2*j + half; if (tap > 8) tap = 8;
    rel[j] = ((tap/3 - 1)*PW + (tap%3 - 1))*2;
  }
  const float* zslot = &WT[144];

  __syncthreads();

  // ---- channel-0 likelihood: mu/log_sigma = b_out broadcast ----
  const float HLOG2PI = 0.9189385332046727f;
  const float mu0 = WT[140], ls0 = WT[141];
  const float e0  = __expf(-ls0);
  float lp = 0.0f;
  {
    const float* x0 = X + (size_t)b * 16 * HW;
    for (int it = 0; it < NIT; ++it) {
      int pix = w*PXW + it*32 + l;
      float z = (x0[pix] - mu0) * e0;
      lp += -0.5f*z*z - ls0 - HLOG2PI;
    }
  }

  // cell state: 32 pixels per thread, 2 features each (private memory)
  float c0r[NIT], c1r[NIT];
#pragma unroll
  for (int i = 0; i < NIT; ++i) { c0r[i] = 0.0f; c1r[i] = 0.0f; }

  float* hold = lds + OFF_H0;
  float* hnew = lds + OFF_H1;
  float* GSw  = lds + OFF_GS + w*288;   // 32 px * stride 9

  for (int t = 0; t < TSTEPS; ++t) {
    const float* xt = X + ((size_t)(b*16 + t)) * HW;

    // ============ phase A: hh-gates via WMMA + LSTM update (reads hold, writes hnew) ========
    for (int it = 0; it < NIT; ++it) {
      const int p0 = w*PXW + it*32;

      // two 16-pixel WMMA groups cover the 32-pixel strip (independent acc chains)
#pragma unroll
      for (int g2 = 0; g2 < 2; ++g2) {
        const int m   = l & 15;
        const int pix = p0 + g2*16 + m;
        const int py  = pix >> 7;
        const int px  = pix & 127;
        const float* base = hold + ((py + 1)*PW + (px + 1))*2;
        v8f acc = {0.0f,0.0f,0.0f,0.0f,0.0f,0.0f,0.0f,0.0f};
#pragma unroll
        for (int j = 0; j < 4; ++j) {   // taps always valid: unconditional ds_load_b64
          v2f a = *(const v2f*)(base + rel[j]);
          acc = __builtin_amdgcn_wmma_f32_16x16x4_f32(
                    false, a, false, barr[j], (short)0, acc, false, false);
        }
        {                               // j==4: address-select zero slot for K>=18 lanes
          const float* p4 = ok4 ? (base + rel[4]) : zslot;
          v2f a = *(const v2f*)p4;
          acc = __builtin_amdgcn_wmma_f32_16x16x4_f32(
                    false, a, false, barr[4], (short)0, acc, false, false);
        }
        // D layout: lane l, vgpr r -> pixel 8*half + r, gate n (= l&15)
        if (n < 8) {
#pragma unroll
          for (int r = 0; r < 8; ++r)
            GSw[(g2*16 + 8*half + r)*9 + n] = acc[r];
        }
      }
      asm volatile("s_wait_dscnt 0" ::: "memory");   // wave-local LDS RAW ordering

      // per-pixel: on-the-fly ig = conv(conv(x,Win),Wih), row-streamed; then LSTM update
      {
        const int pix = p0 + l;
        const int py  = pix >> 7;
        const int px  = pix & 127;

        float zi[9];
#pragma unroll
        for (int q = 0; q < 9; ++q) zi[q] = 0.0f;

#pragma unroll
        for (int r5 = 0; r5 < 5; ++r5) {           // stream 5 x-rows
          const int yy  = py + r5 - 2;
          const bool yok = ((unsigned)yy < 128u);
          const int yyc = yy < 0 ? 0 : (yy > 127 ? 127 : yy);
          float xr[5];
#pragma unroll
          for (int c5 = 0; c5 < 5; ++c5) {         // clamped address + mask (no branches)
            const int xx  = px + c5 - 2;
            const int xxc = xx < 0 ? 0 : (xx > 127 ? 127 : xx);
            float v = xt[yyc*128 + xxc];
            xr[c5] = (yok && (unsigned)xx < 128u) ? v : 0.0f;
          }
#pragma unroll
          for (int qy = -1; qy <= 1; ++qy) {
            const int ry = r5 - 2 - qy;            // Win row index offset
            if (ry >= -1 && ry <= 1) {             // compile-time after unroll
#pragma unroll
              for (int qx = -1; qx <= 1; ++qx)
#pragma unroll
                for (int rx = -1; rx <= 1; ++rx)
                  zi[(qy+1)*3 + (qx+1)] += WT[(ry+1)*3 + (rx+1)] * xr[qx+rx+2];
            }
          }
        }
        const float bin = WT[9];
#pragma unroll
        for (int q = 0; q < 9; ++q) {              // zi SAME-pad: out-of-range centers -> 0
          const int qy = q/3 - 1, qx = q%3 - 1;
          const bool ok = ((unsigned)(py+qy) < 128u) && ((unsigned)(px+qx) < 128u);
          zi[q] = ok ? (zi[q] + bin) : 0.0f;
        }

        float gt[8];
#pragma unroll
        for (int g = 0; g < 8; ++g) {
          float s = GSw[l*9 + g] + WT[88 + g] + WT[96 + g];   // hh-conv + b_ih + b_hh
#pragma unroll
          for (int q = 0; q < 9; ++q) s += WT[16 + q*8 + g] * zi[q];
          gt[g] = s;
        }

        float* hw = hnew + ((py + 1)*PW + (px + 1))*2;
#pragma unroll
        for (int fc = 0; fc < 2; ++fc) {
          float iv = sigf(gt[0 + fc]);
          float gv = tanhf(gt[2 + fc]);
          float fv = sigf(gt[4 + fc] + 1.0f);
          float ov = sigf(gt[6 + fc]);
          float c  = (fc == 0) ? c0r[it] : c1r[it];
          c = fv*c + iv*gv;
          if (fc == 0) c0r[it] = c; else c1r[it] = c;
          hw[fc] = ov * tanhf(c);
        }
      }
    }

    __syncthreads();   // hnew complete for the whole image

    // ============ phase B: output conv (padded, unconditional) + log-prob for channel t+1 ====
    const float* xn = X + ((size_t)(b*16 + t + 1)) * HW;
    for (int it = 0; it < NIT; ++it) {
      const int pix = w*PXW + it*32 + l;
      const int py  = pix >> 7;
      const int px  = pix & 127;
      const float* base = hnew + ((py + 1)*PW + (px + 1))*2;
      float mu = WT[140], lsg = WT[141];
#pragma unroll
      for (int tap = 0; tap < 9; ++tap) {
        const int d = ((tap/3 - 1)*PW + (tap%3 - 1))*2;
        v2f hv = *(const v2f*)(base + d);
        mu  += WT[104 + tap*4 + 0]*hv.x + WT[104 + tap*4 + 2]*hv.y;
        lsg += WT[104 + tap*4 + 1]*hv.x + WT[104 + tap*4 + 3]*hv.y;
      }
      float z = (xn[pix] - mu) * __expf(-lsg);
      lp += -0.5f*z*z - lsg - HLOG2PI;
    }

    // swap buffers (race-free with single per-step barrier)
    float* tmp = hold; hold = hnew; hnew = tmp;
  }

  // ---- block reduction -> out[b] ----
  float* RED = lds + OFF_RED;
  RED[tid] = lp;
  __syncthreads();
  for (int s = NTHREADS/2; s > 0; s >>= 1) {
    if (tid < s) RED[tid] += RED[tid + s];
    __syncthreads();
  }
  if (tid == 0) out[b] = RED[0];
}

extern "C" void kernel_launch(void* const* d_in, const int* in_sizes, int n_in,
                              void* d_out, int out_size, void* d_ws, size_t ws_size,
                              hipStream_t stream) {
  (void)in_sizes; (void)n_in; (void)out_size; (void)d_ws; (void)ws_size;
  const float* X     = (const float*)d_in[0];
  const float* Win   = (const float*)d_in[1];
  const float* b_in  = (const float*)d_in[2];
  const float* Wih   = (const float*)d_in[3];
  const float* b_ih  = (const float*)d_in[4];
  const float* Whh   = (const float*)d_in[5];
  const float* b_hh  = (const float*)d_in[6];
  const float* Wout  = (const float*)d_in[7];
  const float* b_out = (const float*)d_in[8];
  float* out = (float*)d_out;

  // allow >64KB dynamic LDS (CDNA5 workgroup LDS limit is 320KB)
  (void)hipFuncSetAttribute((const void*)clstm_fused_kernel,
                            hipFuncAttributeMaxDynamicSharedMemorySize, LDS_BYTES);

  clstm_fused_kernel<<<dim3(32), dim3(NTHREADS), LDS_BYTES, stream>>>(
      X, Win, b_in, Wih, b_ih, Whh, b_hh, Wout, b_out, out);
}